// DifferentiableBeliefPropagation_62801011802218
// MI455X (gfx1250) — compile-verified
//
#include <hip/hip_runtime.h>
#include <math.h>

// ---------------------------------------------------------------------------
// Chain-CRF forward/backward (belief propagation) for B=16, T=4096, C=128.
//
// Strategy (MI455X / gfx1250):
//  * log-semiring matvec per step rewritten as an FP32 GEMM in the exp domain
//    with a per-row max shift:  a_new = u + m + log(exp(a-m) @ P), where
//    P = exp(log_softmax(log_trans)) is row-stochastic (entries in (0,1]).
//  * Each scan step = 16x128x128 FP32 GEMM -> V_WMMA_F32_16X16X4_F32,
//    8 waves x (32 K-steps of K=4), 4 independent accumulators per wave to
//    keep the WMMA RAW chain at depth 8.
//  * B fragments (P / P^T) are loop-invariant across all 4096 steps -> held
//    in 64 VGPRs per lane for the entire scan (no per-step LDS reads for B,
//    no operand-packing movs). Only the freshly exp'd A matrix round-trips
//    through LDS each step.
//  * Forward and backward scans are independent -> one kernel, grid = 2.
//  * Workload is latency-bound (4096 sequential steps); HBM traffic
//    (~230 MB at 23.3 TB/s) is negligible.
// ---------------------------------------------------------------------------

typedef __attribute__((ext_vector_type(2))) float v2f;
typedef __attribute__((ext_vector_type(8))) float v8f;

#define B_   16
#define T_   4096
#define C_   128
#define ESTR 132   // LDS row stride for E: bank(m,k) = (4m+k)%64 -> conflict-free
                   // b64 A-fragment loads (even k, 8B aligned)

__device__ __forceinline__ float half_shfl_max(float v) {
  // reduce across a 16-lane half-wave (xor offsets < 16 stay in the half)
  #pragma unroll
  for (int off = 1; off < 16; off <<= 1)
    v = fmaxf(v, __shfl_xor(v, off, 32));
  return v;
}

__device__ __forceinline__ v8f wmma_k4(const float* __restrict__ sE,
                                       v2f bfrag, int k0, int hh, int lo,
                                       v8f c) {
  // A: 16x4 f32 fragment. ISA layout: lanes 0-15 M=0..15 {K=k0,k0+1},
  //                                    lanes 16-31 M=0..15 {K=k0+2,k0+3}.
  const int ka = k0 + 2 * hh;
  v2f a;
  a.x = sE[lo * ESTR + ka];
  a.y = sE[lo * ESTR + ka + 1];
  // 8 args: (neg_a, A, neg_b, B, c_mod, C, reuse_a, reuse_b)
  return __builtin_amdgcn_wmma_f32_16x16x4_f32(
      false, a, false, bfrag, (short)0, c, false, false);
}

// ---------------------------------------------------------------------------
// Prep: P = exp(log_softmax(log_trans, rows)); also write P^T.
// ---------------------------------------------------------------------------
__global__ __launch_bounds__(128) void crf_prep(const float* __restrict__ lt,
                                                float* __restrict__ Pg,
                                                float* __restrict__ PTg) {
  const int i = threadIdx.x;  // one row per thread
  float mx = -INFINITY;
  for (int j = 0; j < C_; ++j) mx = fmaxf(mx, lt[i * C_ + j]);
  float s = 0.f;
  for (int j = 0; j < C_; ++j) s += __expf(lt[i * C_ + j] - mx);
  const float inv = 1.0f / s;
  for (int j = 0; j < C_; ++j) {
    const float p = __expf(lt[i * C_ + j] - mx) * inv;
    Pg[i * C_ + j] = p;
    PTg[j * C_ + i] = p;
  }
}

// ---------------------------------------------------------------------------
// Sequential scan (one block = one direction). 256 threads = 8 waves.
//   FWD:  a_t = u_t + m + log(exp(a_{t-1}-m) @ P)              (alpha)
//   BWD:  x = b_{t+1} + u_{t+1}; b_t = m + log(exp(x-m) @ P^T) (beta)
// ---------------------------------------------------------------------------
template <bool FWD>
__device__ void scan_dir(const float* __restrict__ u,
                         const float* __restrict__ mat,  // P or P^T, 128x128
                         float* __restrict__ outbuf,     // alpha or beta (B,T,C)
                         float* smem) {
  float* sE = smem;                 // 16*ESTR exp'd state (A matrix)
  float* sA = sE + B_ * ESTR;       // 16*128 running log-domain state
  float* sM = sA + B_ * C_;         // 16 row maxes

  const int tid  = threadIdx.x;
  const int lane = tid & 31;
  const int wv   = tid >> 5;     // wave id 0..7 -> owns N tile [16w,16w+16)
  const int hh   = lane >> 4;    // half-wave
  const int lo   = lane & 15;
  const int n0   = wv * 16;

  // B fragments of the transition matrix are invariant over the whole scan:
  // keep all 32 K-steps (64 VGPRs) resident in registers. One-time global
  // loads; latency amortized over 4096 iterations.
  v2f bf[32];
  #pragma unroll
  for (int kk = 0; kk < 32; ++kk) {
    const int ka = 4 * kk + 2 * hh;
    bf[kk].x = mat[ka * C_ + n0 + lo];
    bf[kk].y = mat[(ka + 1) * C_ + n0 + lo];
  }

  const int b   = tid >> 4;      // batch row 0..15 (phase-A mapping)
  const int sub = tid & 15;      // 16 threads per batch row
  const int j0  = sub * 8;       // 8 states per thread

  // Boundary condition: alpha_0 = u_0 ; beta_{T-1} = 0.
  {
    const int tEdge = FWD ? 0 : (T_ - 1);
    #pragma unroll
    for (int q = 0; q < 8; ++q) {
      const int j = j0 + q;
      const float v = FWD ? u[(b * T_ + 0) * C_ + j] : 0.0f;
      sA[b * C_ + j] = v;
      outbuf[(b * T_ + tEdge) * C_ + j] = v;
    }
  }
  __syncthreads();

  for (int s = 1; s < T_; ++s) {
    const int t = FWD ? s : (T_ - 1 - s);  // timestep being produced

    // ---- phase A: per-row max + exp into E --------------------------------
    float vals[8];
    float lm = -INFINITY;
    #pragma unroll
    for (int q = 0; q < 8; ++q) {
      float v = sA[b * C_ + j0 + q];
      if (!FWD) v += u[(b * T_ + (t + 1)) * C_ + j0 + q];  // beta_{t+1}+u_{t+1}
      vals[q] = v;
      lm = fmaxf(lm, v);
    }
    lm = half_shfl_max(lm);
    if (sub == 0) sM[b] = lm;
    #pragma unroll
    for (int q = 0; q < 8; ++q)
      sE[b * ESTR + j0 + q] = __expf(vals[q] - lm);

    // Prefetch u_t for phase C (forward) so the global loads are in flight
    // while the WMMA chain executes.
    float uval[8];
    if (FWD) {
      #pragma unroll
      for (int r = 0; r < 8; ++r)
        uval[r] = u[((r + 8 * hh) * T_ + t) * C_ + n0 + lo];
    }
    __syncthreads();

    // ---- phase B: S = E @ P  (16x128x128 fp32 GEMM, WMMA 16x16x4) ---------
    v8f acc0 = {}, acc1 = {}, acc2 = {}, acc3 = {};  // 4 chains of depth 8
    #pragma unroll
    for (int kk = 0; kk < 8; ++kk) {
      acc0 = wmma_k4(sE, bf[4 * kk + 0], 16 * kk + 0,  hh, lo, acc0);
      acc1 = wmma_k4(sE, bf[4 * kk + 1], 16 * kk + 4,  hh, lo, acc1);
      acc2 = wmma_k4(sE, bf[4 * kk + 2], 16 * kk + 8,  hh, lo, acc2);
      acc3 = wmma_k4(sE, bf[4 * kk + 3], 16 * kk + 12, hh, lo, acc3);
    }
    v8f cc = (acc0 + acc1) + (acc2 + acc3);

    // ---- phase C: back to log domain, update state, stream to HBM ---------
    // C layout: VGPR r -> M=r (lanes 0-15) / M=r+8 (lanes 16-31); N = lane&15.
    #pragma unroll
    for (int r = 0; r < 8; ++r) {
      const int M = r + 8 * hh;      // batch row
      const int N = n0 + lo;         // state
      float val = sM[M] + __logf(cc[r]);
      if (FWD) val += uval[r];
      sA[M * C_ + N] = val;
      outbuf[(M * T_ + t) * C_ + N] = val;
    }
    __syncthreads();
  }
}

__global__ __launch_bounds__(256) void crf_scan(const float* __restrict__ u,
                                                const float* __restrict__ Pg,
                                                const float* __restrict__ PTg,
                                                float* __restrict__ alpha,
                                                float* __restrict__ beta) {
  extern __shared__ float smem[];
  if (blockIdx.x == 0)
    scan_dir<true>(u, Pg, alpha, smem);
  else
    scan_dir<false>(u, PTg, beta, smem);
}

// ---------------------------------------------------------------------------
// Finalize: out = log_softmax(alpha + beta, axis=-1). One 128-thread block
// per (b,t) row; alpha/beta/out share the (B,T,C) flat layout.
// ---------------------------------------------------------------------------
__global__ __launch_bounds__(128) void crf_finalize(
    const float* __restrict__ alpha, const float* __restrict__ beta,
    float* __restrict__ out) {
  __shared__ float redm[4];
  __shared__ float reds[4];
  const int row = blockIdx.x;       // 0..B*T-1
  const int c   = threadIdx.x;      // 0..127
  const float v = alpha[row * C_ + c] + beta[row * C_ + c];

  float m = v;
  #pragma unroll
  for (int off = 1; off < 32; off <<= 1) m = fmaxf(m, __shfl_xor(m, off, 32));
  const int wv = c >> 5;
  if ((c & 31) == 0) redm[wv] = m;
  __syncthreads();
  m = fmaxf(fmaxf(redm[0], redm[1]), fmaxf(redm[2], redm[3]));

  float e = __expf(v - m);
  #pragma unroll
  for (int off = 1; off < 32; off <<= 1) e += __shfl_xor(e, off, 32);
  if ((c & 31) == 0) reds[wv] = e;
  __syncthreads();
  const float s = reds[0] + reds[1] + reds[2] + reds[3];

  out[row * C_ + c] = v - m - __logf(s);
}

// ---------------------------------------------------------------------------
extern "C" void kernel_launch(void* const* d_in, const int* in_sizes, int n_in,
                              void* d_out, int out_size, void* d_ws,
                              size_t ws_size, hipStream_t stream) {
  (void)in_sizes; (void)n_in; (void)out_size; (void)ws_size;
  const float* u  = (const float*)d_in[0];   // (B,T,C) fp32
  const float* lt = (const float*)d_in[1];   // (C,C)   fp32
  float* out = (float*)d_out;                // (B,T,C) fp32

  // Workspace layout: P | P^T | alpha | beta  (~64.1 MB)
  float* ws    = (float*)d_ws;
  float* Pg    = ws;
  float* PTg   = Pg + (size_t)C_ * C_;
  float* alpha = PTg + (size_t)C_ * C_;
  float* beta  = alpha + (size_t)B_ * T_ * C_;

  crf_prep<<<1, 128, 0, stream>>>(lt, Pg, PTg);

  const size_t smemBytes =
      (size_t)(B_ * ESTR + B_ * C_ + 16) * sizeof(float);  // ~16.7 KB
  crf_scan<<<2, 256, smemBytes, stream>>>(u, Pg, PTg, alpha, beta);

  crf_finalize<<<B_ * T_, 128, 0, stream>>>(alpha, beta, out);
}